// WeightedMessageAggregator_80101140070794
// MI455X (gfx1250) — compile-verified
//
#include <hip/hip_runtime.h>
#include <hip/hip_bf16.h>
#include <climits>

typedef __attribute__((ext_vector_type(2))) float v2f;
typedef __attribute__((ext_vector_type(8))) float v8f;

static constexpr int kD = 128;          // message dim
static constexpr int kWaves = 8;        // waves per block (8 * 32 = 256 threads)

// ---------------------------------------------------------------------------
// Kernel 1: segment offsets via binary search over sorted seg_ids.
// seg_start[n] = lower_bound(seg_ids, n). seg_ids is 8MB -> L2 resident.
// ---------------------------------------------------------------------------
__global__ void seg_bounds_kernel(const int* __restrict__ seg_ids, int M, int N,
                                  int* __restrict__ seg_start) {
  int n = blockIdx.x * blockDim.x + threadIdx.x;
  if (n >= N) return;
  int lo = 0, hi = M;
  while (lo < hi) {
    int mid = (lo + hi) >> 1;
    if (seg_ids[mid] < n) lo = mid + 1; else hi = mid;
  }
  seg_start[n] = lo;
}

// ---------------------------------------------------------------------------
// Kernel 2: fused per-segment online softmax + weighted aggregation.
// One wave32 per segment. Single pass over msgs (1.02 GB total read).
// Weighted accumulation runs on V_WMMA_F32_16X16X4_F32:
//   acc_b(16ch x 16) += A(16ch x 4msgs) * E(4msgs x 16),  E[k][c] = e_k
// A layout (ISA 7.12.2, 32-bit 16x4): lanes 0-15 = K0(vgpr0),K1(vgpr1);
// lanes 16-31 = K2,K3.  B mirrors the K split: vgpr0 = {e0 | e2}, vgpr1 = {e1 | e3}.
// ---------------------------------------------------------------------------
__global__ void __launch_bounds__(kWaves * 32)
seg_softmax_agg_kernel(const float* __restrict__ msgs,
                       const int* __restrict__ timestamps,
                       const float* __restrict__ W,
                       const float* __restrict__ bias_p,
                       const int* __restrict__ seg_start,
                       int M, int N,
                       float* __restrict__ out_agg,
                       int* __restrict__ out_ts) {
  __shared__ float lds[kWaves * kD];

  const int lane = threadIdx.x & 31;
  const int wave = threadIdx.x >> 5;
  const int col  = lane & 15;   // WMMA column / A-row index
  const int half = lane >> 4;   // 0: K=0,1   1: K=2,3
  const int n = blockIdx.x * kWaves + wave;

  float denom = 0.0f;
  int m = 0;

  if (n < N) {
    const float bias = bias_p[0];
    const int start = seg_start[n];
    const int end   = (n + 1 < N) ? seg_start[n + 1] : M;
    m = end - start;

    // W cached per lane: wv[b] = W[b*16 + col]
    float wv[8];
#pragma unroll
    for (int b = 0; b < 8; ++b) wv[b] = W[b * 16 + col];

    v8f acc[8] = {};          // 8 channel-blocks of 16, D-tiles of the WMMA
    float run_max = -INFINITY;

    for (int base = 0; base < m; base += 4) {
      const int j0 = start + base;
      int r0 = j0 + 2 * half;
      int r1 = r0 + 1;
      r0 = min(r0, end - 1);                 // tail clamp (weights zeroed below)
      r1 = min(r1, end - 1);
      const float* p0 = msgs + (long)r0 * kD + col;
      const float* p1 = msgs + (long)r1 * kD + col;
      __builtin_prefetch(p0 + 4 * kD, 0, 0); // global_prefetch_b8 read-ahead

      // Load 4 message rows in A-matrix layout: a0 = K(0|2), a1 = K(1|3)
      float a0[8], a1[8];
#pragma unroll
      for (int b = 0; b < 8; ++b) { a0[b] = p0[b * 16]; a1[b] = p1[b * 16]; }

      // Scores: per-lane partial dot, butterfly-reduce within 16-lane halves
      float s0p = 0.0f, s1p = 0.0f;
#pragma unroll
      for (int b = 0; b < 8; ++b) { s0p += a0[b] * wv[b]; s1p += a1[b] * wv[b]; }
#pragma unroll
      for (int off = 1; off <= 8; off <<= 1) {
        s0p += __shfl_xor(s0p, off, 32);
        s1p += __shfl_xor(s1p, off, 32);
      }
      const float t0 = __shfl_xor(s0p, 16, 32);
      const float t1 = __shfl_xor(s1p, 16, 32);
      float s0 = ((half == 0) ? s0p : t0) + bias;
      float s1 = ((half == 0) ? s1p : t1) + bias;
      float s2 = ((half == 0) ? t0 : s0p) + bias;
      float s3 = ((half == 0) ? t1 : s1p) + bias;
      if (base + 1 >= m) s1 = -INFINITY;     // tail messages get weight 0
      if (base + 2 >= m) s2 = -INFINITY;
      if (base + 3 >= m) s3 = -INFINITY;

      // Online softmax (wave-uniform scalars -> uniform branch)
      const float cmax = fmaxf(fmaxf(s0, s1), fmaxf(s2, s3));
      if (cmax > run_max) {
        const float scale = __expf(run_max - cmax);  // 0 on first chunk
        denom *= scale;
#pragma unroll
        for (int b = 0; b < 8; ++b) acc[b] *= scale;
        run_max = cmax;
      }
      const float e0 = __expf(s0 - run_max);
      const float e1 = __expf(s1 - run_max);
      const float e2 = __expf(s2 - run_max);
      const float e3 = __expf(s3 - run_max);
      denom += (e0 + e1) + (e2 + e3);

      // E matrix (B operand): every column identical, rows = e_k
      v2f B;
      B.x = half ? e2 : e0;
      B.y = half ? e3 : e1;

      // acc_b += A_b(16x4) x E(4x16)  -- 8x v_wmma_f32_16x16x4_f32
#pragma unroll
      for (int b = 0; b < 8; ++b) {
        v2f A; A.x = a0[b]; A.y = a1[b];
        acc[b] = __builtin_amdgcn_wmma_f32_16x16x4_f32(
            /*neg_a=*/false, A, /*neg_b=*/false, B,
            /*c_mod=*/(short)0, acc[b], /*reuse_a=*/false, /*reuse_b=*/false);
      }
    }

    // Stage D-tile to LDS. D layout: vgpr v holds row M=v (lanes 0-15) /
    // M=v+8 (lanes 16-31); all 16 columns identical -> replicated same-value
    // writes are safe.
    float* slot = lds + wave * kD;
#pragma unroll
    for (int b = 0; b < 8; ++b) {
#pragma unroll
      for (int v = 0; v < 8; ++v) {
        slot[b * 16 + v + 8 * half] = acc[b][v];
      }
    }
  }

  __syncthreads();

  if (n < N) {
    const float inv = (m > 0) ? (1.0f / denom) : 0.0f;  // empty segment -> 0s
    const float* slot = lds + wave * kD;
    float4 r;
    r.x = slot[4 * lane + 0] * inv;
    r.y = slot[4 * lane + 1] * inv;
    r.z = slot[4 * lane + 2] * inv;
    r.w = slot[4 * lane + 3] * inv;
    *(float4*)(out_agg + (long)n * kD + 4 * lane) = r;   // coalesced 512B row

    // Timestamp segment-max (== last, ids sorted & time-ordered per segment)
    const int start = seg_start[n];
    const int end   = (n + 1 < N) ? seg_start[n + 1] : M;
    int tmax = INT_MIN;
    for (int i = start + lane; i < end; i += 32) tmax = max(tmax, timestamps[i]);
#pragma unroll
    for (int off = 1; off <= 16; off <<= 1) tmax = max(tmax, __shfl_xor(tmax, off, 32));
    if (lane == 0) out_ts[n] = tmax;                     // empty -> INT32_MIN
  }
}

// ---------------------------------------------------------------------------
// Inputs (setup_inputs order): msgs[M*128] f32, seg_ids[M] i32,
// timestamps[M] i32, W[128] f32, b[1] f32, num_segments[1] i32.
// Output: aggregated[N*128] f32 then ts[N] i32, concatenated.
// ---------------------------------------------------------------------------
extern "C" void kernel_launch(void* const* d_in, const int* in_sizes, int n_in,
                              void* d_out, int out_size, void* d_ws, size_t ws_size,
                              hipStream_t stream) {
  const float* msgs       = (const float*)d_in[0];
  const int*   seg_ids    = (const int*)d_in[1];
  const int*   timestamps = (const int*)d_in[2];
  const float* W          = (const float*)d_in[3];
  const float* bias       = (const float*)d_in[4];

  const int M = in_sizes[1];
  const int N = out_size / (kD + 1);     // out = N*128 floats + N ints

  int*   seg_start = (int*)d_ws;
  float* out_agg   = (float*)d_out;
  int*   out_ts    = (int*)((float*)d_out + (size_t)N * kD);

  seg_bounds_kernel<<<(N + 255) / 256, 256, 0, stream>>>(seg_ids, M, N, seg_start);

  const int blocks = (N + kWaves - 1) / kWaves;
  seg_softmax_agg_kernel<<<blocks, kWaves * 32, 0, stream>>>(
      msgs, timestamps, W, bias, seg_start, M, N, out_agg, out_ts);
}